// EventSpace_4552665333774
// MI455X (gfx1250) — compile-verified
//
#include <hip/hip_runtime.h>

// ---- shapes (compile-time constants from the reference) ----
// B=4, S=64, D=64, UNITS=64, ROUTINGS=3
// space: (4, 4096, 4096) f32 = 256MB ; out tuple = [outputs(4,64,64), updated_space(4,4096,4096)]

typedef float v2f __attribute__((ext_vector_type(2)));
typedef float v8f __attribute__((ext_vector_type(8)));
typedef float f4  __attribute__((ext_vector_type(4)));

// workspace layout (floats):
//   [0,      16384)  cross  (B,64,64)
//   [16384,  16388)  reward_diff[b] * (1/sqrt(2))
//   [16448,  16448+1048576)  u_hat  (B,[n=64][i=64][d=64])
#define WS_CROSS 0
#define WS_RD    16384
#define WS_UHAT  16448
#define UHAT_B   262144
#define SP_B     16777216ull   // 4096*4096

// -------- kernel 0: diag gather + leaky-relu cross, reward_diff --------
__global__ void k_prep(const float* __restrict__ inputs,
                       const float* __restrict__ space,
                       const float* __restrict__ bias,
                       float* __restrict__ ws)
{
    int b = blockIdx.x;
    int t = threadIdx.x;
    const float* inB = inputs + b * 4096;
    const float* spB = space + (size_t)b * SP_B;
    float* crossB = ws + WS_CROSS + b * 4096;

    // cross[j,k] = leaky( inputs[b,j,k] * space[b, j*65, k*65] )
    for (int idx = t; idx < 4096; idx += 256) {
        int j = idx >> 6, k = idx & 63;
        float dg = spB[(size_t)(j * 65) * 4096 + (size_t)(k * 65)];
        float v = inB[idx] * dg;
        crossB[idx] = v >= 0.0f ? v : 0.2f * v;
    }

    // reward_diff[b] = tanh( max_s |space[b,0,4095] - inputs[b,s,63]| - bias[b] )
    __shared__ float red[64];
    float ideal00 = spB[4095];
    if (t < 64) red[t] = fabsf(ideal00 - inB[t * 64 + 63]);
    __syncthreads();
    if (t == 0) {
        float m = red[0];
        for (int s = 1; s < 64; ++s) m = fmaxf(m, red[s]);
        float rd = tanhf(m - bias[b]);
        ws[WS_RD + b] = rd * 0.70710678118654752440f;  // fold 1/sqrt(2)
    }
}

// -------- kernel 1: u_hat = cross(64x64) @ W_caps(64x4096) via f32 WMMA --------
// One 16x16 output tile per wave32; K accumulated in 16 steps of 4.
// Stored transposed into ws as u_hat[b][n][i][d] (n = e>>6, d = e&63).
__global__ void k_uhat_wmma(const float* __restrict__ Wc,
                            float* __restrict__ ws)
{
    int wave = threadIdx.x >> 5;
    int lane = threadIdx.x & 31;
    int tile = blockIdx.x * 4 + wave;        // 4096 tiles total
    int b    = tile >> 10;
    int rem  = tile & 1023;
    int m0   = (rem >> 8) << 4;              // 4 row tiles
    int n0   = (rem & 255) << 4;             // 256 col tiles

    const float* crossB = ws + WS_CROSS + b * 4096;
    float* uhat = ws + WS_UHAT + b * UHAT_B;

    int half = lane >> 4;                    // K sub-slot per ISA A/B layout
    int lr   = lane & 15;

    const float* arow = crossB + (m0 + lr) * 64;       // A: row m0+lr
    const float* bcol = Wc + (n0 + lr);                // B: col n0+lr

    v8f acc = {0.f, 0.f, 0.f, 0.f, 0.f, 0.f, 0.f, 0.f};
    #pragma unroll
    for (int k0 = 0; k0 < 64; k0 += 4) {
        int ka = k0 + half * 2;
        v2f a;  a.x  = arow[ka];                 a.y  = arow[ka + 1];
        v2f bb; bb.x = bcol[(size_t)ka * 4096];  bb.y = bcol[(size_t)(ka + 1) * 4096];
        acc = __builtin_amdgcn_wmma_f32_16x16x4_f32(
            /*neg_a=*/false, a, /*neg_b=*/false, bb,
            /*c_mod=*/(short)0, acc, /*reuse_a=*/false, /*reuse_b=*/false);
    }

    // C/D layout: VGPR v, lane L: M = v + 8*(L>>4), N = L&15
    int e = n0 + lr;
    int n = e >> 6, d = e & 63;
    float* dst = uhat + n * 4096 + d;
    #pragma unroll
    for (int v = 0; v < 8; ++v) {
        int srow = m0 + v + half * 8;
        dst[srow * 64] = acc[v];
    }
}

// -------- kernel 2: dynamic routing (3 iters) + sequence encoder --------
__global__ void k_route(const float* __restrict__ Wt,
                        const float* __restrict__ Wf,
                        const float* __restrict__ benc,
                        const float* __restrict__ ws,
                        float* __restrict__ out)
{
    __shared__ float bl[4096];   // b logits [n][i]
    __shared__ float cs[4096];   // c [n][i]; reused as tmp [s][v] for encoder
    __shared__ float os[4096];   // o [n][d]

    int b = blockIdx.x;
    int t = threadIdx.x;
    const float* Uh = ws + WS_UHAT + b * UHAT_B;

    for (int idx = t; idx < 4096; idx += 256) bl[idx] = 0.0f;

    for (int it = 0; it < 3; ++it) {
        __syncthreads();
        if (t < 64) {                               // softmax over n, per column i
            int i = t;
            float m = bl[i];
            for (int n = 1; n < 64; ++n) m = fmaxf(m, bl[n * 64 + i]);
            float ssum = 0.0f;
            for (int n = 0; n < 64; ++n) ssum += expf(bl[n * 64 + i] - m);
            float inv = 1.0f / ssum;
            for (int n = 0; n < 64; ++n) cs[n * 64 + i] = expf(bl[n * 64 + i] - m) * inv;
        }
        __syncthreads();
        {                                           // o[n,d] = sum_i c[n,i]*u_hat[n,i,d]
            int n  = t >> 2;
            int db = (t & 3) << 4;
            float acc[16];
            #pragma unroll
            for (int dd = 0; dd < 16; ++dd) acc[dd] = 0.0f;
            const float* Un = Uh + n * 4096 + db;
            for (int i = 0; i < 64; ++i) {
                float cc = cs[n * 64 + i];
                const float* r = Un + i * 64;
                #pragma unroll
                for (int dd = 0; dd < 16; ++dd) acc[dd] += cc * r[dd];
            }
            #pragma unroll
            for (int dd = 0; dd < 16; ++dd) os[n * 64 + db + dd] = acc[dd];
        }
        __syncthreads();
        if (t < 64) {                               // squash over d, per n
            int n = t;
            float ssum = 0.0f;
            for (int dd = 0; dd < 64; ++dd) { float v = os[n * 64 + dd]; ssum += v * v; }
            float sc = (ssum / (1.0f + ssum)) * rsqrtf(ssum + 1e-7f);
            for (int dd = 0; dd < 64; ++dd) os[n * 64 + dd] *= sc;
        }
        __syncthreads();
        if (it < 2) {                               // b[n,i] += sum_d o[n,d]*u_hat[n,i,d]
            int base = t * 16;
            for (int r = 0; r < 16; ++r) {
                int ni = base + r;
                int n = ni >> 6, i = ni & 63;
                const float* ur   = Uh + n * 4096 + i * 64;
                const float* orow = os + n * 64;
                float acc = 0.0f;
                for (int dd = 0; dd < 64; ++dd) acc += orow[dd] * ur[dd];
                bl[ni] += acc;
            }
        }
    }
    __syncthreads();

    // encoder: tmp[s,v] = sum_u W_time[u,s]*o[u,v]   (reuse cs for tmp)
    {
        int s = t >> 2, vb = (t & 3) << 4;
        float acc[16];
        #pragma unroll
        for (int v = 0; v < 16; ++v) acc[v] = 0.0f;
        for (int u = 0; u < 64; ++u) {
            float wt = Wt[u * 64 + s];
            const float* orow = os + u * 64 + vb;
            #pragma unroll
            for (int v = 0; v < 16; ++v) acc[v] += wt * orow[v];
        }
        #pragma unroll
        for (int v = 0; v < 16; ++v) cs[s * 64 + vb + v] = acc[v];
    }
    __syncthreads();
    // out[s,d] = relu( sum_v tmp[s,v]*W_feat[v,d] + b_enc[s,d] )
    {
        int s = t >> 2, db = (t & 3) << 4;
        float acc[16];
        #pragma unroll
        for (int dd = 0; dd < 16; ++dd) acc[dd] = 0.0f;
        for (int v = 0; v < 64; ++v) {
            float tv = cs[s * 64 + v];
            const float* wf = Wf + v * 64 + db;
            #pragma unroll
            for (int dd = 0; dd < 16; ++dd) acc[dd] += tv * wf[dd];
        }
        float* od = out + b * 4096 + s * 64 + db;
        const float* be = benc + s * 64 + db;
        #pragma unroll
        for (int dd = 0; dd < 16; ++dd) od[dd] = fmaxf(acc[dd] + be[dd], 0.0f);
    }
}

// -------- kernel 3: 512MB streaming space update (the roofline kernel) --------
// upd[b, i*64+p, j*64+q] = space[...] + rd[b]/sqrt2 * out[b,i,j] * in[b,p,q]
// One block per (b, i*64+p) row of 4096 floats; float4 NT read/modify/write.
__global__ void k_space(const float* __restrict__ inputs,
                        const float* __restrict__ space,
                        const float* __restrict__ outs,
                        const float* __restrict__ ws,
                        float* __restrict__ upd)
{
    __shared__ float orow[64];
    __shared__ float irow[64];
    int blk = blockIdx.x;          // = b*4096 + r
    int b = blk >> 12;
    int r = blk & 4095;
    int i = r >> 6, p = r & 63;
    int t = threadIdx.x;

    if (t < 64)        orow[t]      = outs[b * 4096 + i * 64 + t];
    else if (t < 128)  irow[t - 64] = inputs[b * 4096 + p * 64 + (t - 64)];
    __syncthreads();

    float coeff = ws[WS_RD + b];               // rd[b] / sqrt(2)
    int q0 = (t & 15) << 2;
    f4 in4 = *(const f4*)&irow[q0];
    int jb = t >> 4;
    size_t base = (size_t)blk * 4096 + (size_t)(t << 2);

    #pragma unroll
    for (int u = 0; u < 4; ++u) {              // c = t*4 + u*1024, fully coalesced 128b
        float sc = coeff * orow[jb + (u << 4)];
        f4 s4 = __builtin_nontemporal_load((const f4*)(space + base + u * 1024));
        f4 o4 = s4 + sc * in4;
        __builtin_nontemporal_store(o4, (f4*)(upd + base + u * 1024));
    }
}

extern "C" void kernel_launch(void* const* d_in, const int* in_sizes, int n_in,
                              void* d_out, int out_size, void* d_ws, size_t ws_size,
                              hipStream_t stream)
{
    const float* inputs = (const float*)d_in[0];
    const float* space  = (const float*)d_in[1];
    const float* bias   = (const float*)d_in[2];
    const float* Wc     = (const float*)d_in[3];
    const float* Wt     = (const float*)d_in[4];
    const float* Wf     = (const float*)d_in[5];
    const float* benc   = (const float*)d_in[6];
    float* out = (float*)d_out;                 // outputs (16384 floats) first
    float* upd = out + 16384;                   // then updated_space (67108864 floats)
    float* ws  = (float*)d_ws;

    k_prep     <<<4,     256, 0, stream>>>(inputs, space, bias, ws);
    k_uhat_wmma<<<1024,  128, 0, stream>>>(Wc, ws);
    k_route    <<<4,     256, 0, stream>>>(Wt, Wf, benc, ws, out);
    k_space    <<<16384, 256, 0, stream>>>(inputs, space, out, ws, upd);
}